// SpectralConv2d_48962627174525
// MI455X (gfx1250) — compile-verified
//
#include <hip/hip_runtime.h>

typedef __attribute__((ext_vector_type(2))) float v2f;
typedef __attribute__((ext_vector_type(8))) float v8f;

// D(16x16,f32) = A(16x4,f32) * B(4x16,f32) + C   -- V_WMMA_F32_16X16X4_F32
static __device__ __forceinline__ v8f wmma4(v2f a, v2f b, v8f c) {
  return __builtin_amdgcn_wmma_f32_16x16x4_f32(
      /*neg_a=*/false, a, /*neg_b=*/false, b,
      /*c_mod=*/(short)0, c, /*reuse_a=*/false, /*reuse_b=*/false);
}

#define TWO_PI_OVER_256 0.0245436926061702596f

// ---------------------------------------------------------------------------
// Stage A: per (b,ci) image, the 16x16 low-mode rfft2 coefficients.
//   M1  = x * Fw        (256x256)*(256x16 cplx)   Fw = e^{-2pi i w kw/256}
//   xft = Fh^T * M1     (16x256 cplx)*(256x16 cplx)
// grid = 512 (B*Ci), block = 256 (8 waves). Output: xft[img][kh*16+kw][2].
// ---------------------------------------------------------------------------
__global__ __launch_bounds__(256) void fno_fwd_dft(const float* __restrict__ x,
                                                   float* __restrict__ xft) {
  __shared__ float cs[256], sn[256], sneg[256];
  __shared__ float m1r[256 * 16], m1i[256 * 16];
  __shared__ float acc[16 * 16 * 2];

  const int tid  = threadIdx.x;
  const int lane = tid & 31;
  const int wv   = tid >> 5;
  const int half = lane >> 4;   // 0: lanes 0-15, 1: lanes 16-31
  const int ml   = lane & 15;

  {
    float a = (float)tid * TWO_PI_OVER_256;
    float c = cosf(a), s = sinf(a);
    cs[tid] = c;  sn[tid] = s;  sneg[tid] = -s;
    acc[2 * tid + 0] = 0.0f;
    acc[2 * tid + 1] = 0.0f;
  }
  __syncthreads();

  const int img = blockIdx.x;                 // b*32 + ci
  const float* xim = x + (size_t)img * 256 * 256;

  // ---- stage 1: M1 = x * Fw. Each wave: 2 row-tiles, twiddles shared. ----
  const int rt0 = wv * 2, rt1 = rt0 + 1;
  const float* xr0 = xim + (size_t)(rt0 * 16 + ml) * 256 + 2 * half;
  const float* xr1 = xim + (size_t)(rt1 * 16 + ml) * 256 + 2 * half;
  v8f ar0 = {}, ai0 = {}, ar1 = {}, ai1 = {};
  int ib = (2 * half) * ml;                   // running (k * ml) mod 256
  for (int k0 = 0; k0 < 256; k0 += 4) {
    const int i0 = ib & 255;
    const int i1 = (ib + ml) & 255;
    v2f br, bi;                               // B[k][n=ml] = e^{-i 2pi k*ml/256}
    br.x = cs[i0];   br.y = cs[i1];
    bi.x = sneg[i0]; bi.y = sneg[i1];
    v2f a0, a1;                               // A[m=ml][k]
    a0.x = xr0[k0]; a0.y = xr0[k0 + 1];
    a1.x = xr1[k0]; a1.y = xr1[k0 + 1];
    ar0 = wmma4(a0, br, ar0);  ai0 = wmma4(a0, bi, ai0);
    ar1 = wmma4(a1, br, ar1);  ai1 = wmma4(a1, bi, ai1);
    ib += 4 * ml;
  }
#pragma unroll
  for (int v = 0; v < 8; ++v) {
    const int r0 = rt0 * 16 + v + 8 * half;   // C/D layout: VGPR v -> M=v / v+8
    const int r1 = rt1 * 16 + v + 8 * half;
    m1r[r0 * 16 + ml] = ar0[v];  m1i[r0 * 16 + ml] = ai0[v];
    m1r[r1 * 16 + ml] = ar1[v];  m1i[r1 * 16 + ml] = ai1[v];
  }
  __syncthreads();

  // ---- stage 2: xft = Fh^T * M1, K=256 split across 8 waves (32 each) ----
  v8f xr = {}, xi = {};
  int ib2 = ml * (wv * 32 + 2 * half);        // running (kh * h) mod 256
  for (int k0 = wv * 32; k0 < wv * 32 + 32; k0 += 4) {
    const int ka = k0 + 2 * half;
    const int i0 = ib2 & 255;
    const int i1 = (ib2 + ml) & 255;
    v2f ac, as, ans;                          // A[m=kh=ml][k=h] twiddles
    ac.x  = cs[i0];    ac.y  = cs[i1];
    as.x  = sn[i0];    as.y  = sn[i1];
    ans.x = sneg[i0];  ans.y = sneg[i1];
    v2f br, bi;                               // B[k=h][n=kw=ml] from LDS M1
    br.x = m1r[ka * 16 + ml];  br.y = m1r[(ka + 1) * 16 + ml];
    bi.x = m1i[ka * 16 + ml];  bi.y = m1i[(ka + 1) * 16 + ml];
    // xft_r = C^T M1r + S^T M1i ; xft_i = C^T M1i - S^T M1r
    xr = wmma4(ac, br, xr);   xr = wmma4(as, bi, xr);
    xi = wmma4(ac, bi, xi);   xi = wmma4(ans, br, xi);
    ib2 += 4 * ml;
  }
#pragma unroll
  for (int v = 0; v < 8; ++v) {
    const int kh = v + 8 * half;
    atomicAdd(&acc[(kh * 16 + ml) * 2 + 0], xr[v]);  // ds_add_f32
    atomicAdd(&acc[(kh * 16 + ml) * 2 + 1], xi[v]);
  }
  __syncthreads();

  xft[(size_t)img * 512 + tid]       = acc[tid];
  xft[(size_t)img * 512 + 256 + tid] = acc[256 + tid];
}

// ---------------------------------------------------------------------------
// Stage B: per-mode complex channel mixing. low[b,co] = sum_ci xft[b,ci]*W[ci,co]
// grid = 256 modes, block = 32 (1 wave). M=16(batch), N=32(co), K=32(ci).
// A-fragments shared across the two N-tiles.
// ---------------------------------------------------------------------------
__global__ __launch_bounds__(32) void fno_mix(const float* __restrict__ xft,
                                              const float* __restrict__ wts,
                                              float* __restrict__ low) {
  const int mode = blockIdx.x;                // kh*16 + kw
  const int lane = threadIdx.x & 31;
  const int half = lane >> 4;
  const int ml   = lane & 15;

  v8f accr0 = {}, acci0 = {}, accr1 = {}, acci1 = {};
  const int n0 = ml, n1 = 16 + ml;            // co columns for the two N-tiles

  for (int k0 = 0; k0 < 32; k0 += 4) {
    const int ka = k0 + 2 * half;
    // A[m=b=ml][k=ci]: xft layout ((b*32+ci)*256 + mode)*2 + {r,i}
    const float* pa0 = xft + ((size_t)(ml * 32 + ka) * 256 + mode) * 2;
    const float* pa1 = xft + ((size_t)(ml * 32 + ka + 1) * 256 + mode) * 2;
    v2f arf, aif;
    arf.x = pa0[0]; arf.y = pa1[0];
    aif.x = pa0[1]; aif.y = pa1[1];
    // B[k=ci][n=co]: weights flat = ci*16384 + co*512 + mode*2 + {r,i}
    const float* pb00 = wts + (size_t)ka * 16384 + (size_t)n0 * 512 + mode * 2;
    const float* pb01 = pb00 + 16384;
    const float* pb10 = wts + (size_t)ka * 16384 + (size_t)n1 * 512 + mode * 2;
    const float* pb11 = pb10 + 16384;
    v2f br0, bi0, bn0, br1, bi1, bn1;
    br0.x = pb00[0]; br0.y = pb01[0];
    bi0.x = pb00[1]; bi0.y = pb01[1];
    bn0.x = -bi0.x;  bn0.y = -bi0.y;
    br1.x = pb10[0]; br1.y = pb11[0];
    bi1.x = pb10[1]; bi1.y = pb11[1];
    bn1.x = -bi1.x;  bn1.y = -bi1.y;
    // complex GEMM: r = Ar*Wr - Ai*Wi ; i = Ar*Wi + Ai*Wr
    accr0 = wmma4(arf, br0, accr0);  accr0 = wmma4(aif, bn0, accr0);
    acci0 = wmma4(arf, bi0, acci0);  acci0 = wmma4(aif, br0, acci0);
    accr1 = wmma4(arf, br1, accr1);  accr1 = wmma4(aif, bn1, accr1);
    acci1 = wmma4(arf, bi1, acci1);  acci1 = wmma4(aif, br1, acci1);
  }
#pragma unroll
  for (int v = 0; v < 8; ++v) {
    const int b = v + 8 * half;
    float* p0 = low + ((size_t)(b * 32 + ml) * 256 + mode) * 2;
    float* p1 = low + ((size_t)(b * 32 + 16 + ml) * 256 + mode) * 2;
    p0[0] = accr0[v];  p0[1] = acci0[v];
    p1[0] = accr1[v];  p1[1] = acci1[v];
  }
}

// ---------------------------------------------------------------------------
// Stage C: per (b,co) image inverse transform.
//   G   = E * low           E[h][kh] = e^{+2pi i kh h/256}   (256x16 cplx)
//   out = Gr*C2 + Gi*S2
// C2/S2 tables carry 2/(H*W) and the -sin; the kw=0 Hermitian alpha=1 case is
// folded in by halving G's kw=0 column when staging.
// grid = 512 (B*Co), block = 256 (8 waves, 2 row-tiles each).
// ---------------------------------------------------------------------------
__global__ __launch_bounds__(256) void fno_inv(const float* __restrict__ low,
                                               float* __restrict__ out) {
  __shared__ float cs[256], sn[256], sneg[256], c2[256], s2[256];
  __shared__ float lr[256], li[256];           // 16x16 low tile
  __shared__ float gst[8][4][16 * 16];         // per-wave {Gr0,Gi0,Gr1,Gi1}

  const int tid  = threadIdx.x;
  const int lane = tid & 31;
  const int wv   = tid >> 5;
  const int half = lane >> 4;
  const int ml   = lane & 15;
  const int img  = blockIdx.x;                 // b*32 + co

  {
    float a = (float)tid * TWO_PI_OVER_256;
    float c = cosf(a), s = sinf(a);
    const float k2 = 2.0f / 65536.0f;          // 2 / (H*W)
    cs[tid] = c;  sn[tid] = s;  sneg[tid] = -s;
    c2[tid] = k2 * c;  s2[tid] = -k2 * s;
    const float* lp = low + (size_t)img * 512;
    lr[tid] = lp[2 * tid + 0];
    li[tid] = lp[2 * tid + 1];
  }
  __syncthreads();

  float* outim = out + (size_t)img * 65536;

  // ---- G tiles for both row-tiles of this wave ----
#pragma unroll
  for (int t = 0; t < 2; ++t) {
    const int rt = wv * 2 + t;
    const int h  = rt * 16 + ml;
    v8f gr = {}, gi = {};
    int ih = h * 2 * half;                     // running (h * kh) mod 256
    for (int k0 = 0; k0 < 16; k0 += 4) {
      const int ka = k0 + 2 * half;
      const int i0 = ih & 255;
      const int i1 = (ih + h) & 255;
      v2f ac, as, ans;
      ac.x  = cs[i0];    ac.y  = cs[i1];
      as.x  = sn[i0];    as.y  = sn[i1];
      ans.x = sneg[i0];  ans.y = sneg[i1];
      v2f br, bi;
      br.x = lr[ka * 16 + ml];  br.y = lr[(ka + 1) * 16 + ml];
      bi.x = li[ka * 16 + ml];  bi.y = li[(ka + 1) * 16 + ml];
      // Gr = C*lr - S*li ; Gi = C*li + S*lr
      gr = wmma4(ac, br, gr);   gr = wmma4(ans, bi, gr);
      gi = wmma4(ac, bi, gi);   gi = wmma4(as, br, gi);
      ih += 4 * h;
    }
    const float f = (ml == 0) ? 0.5f : 1.0f;   // fold alpha_kw=0 into G
#pragma unroll
    for (int v = 0; v < 8; ++v) {
      const int row = v + 8 * half;
      gst[wv][2 * t + 0][row * 16 + ml] = gr[v] * f;
      gst[wv][2 * t + 1][row * 16 + ml] = gi[v] * f;
    }
  }
  __syncthreads();  // uniform; each wave reads only its own staging area

  // ---- preload all A-fragments for GEMM2 (invariant over N-tiles) ----
  v2f agr0[4], agi0[4], agr1[4], agi1[4];
#pragma unroll
  for (int ks = 0; ks < 4; ++ks) {
    const int ka = 4 * ks + 2 * half;
    agr0[ks].x = gst[wv][0][ml * 16 + ka];  agr0[ks].y = gst[wv][0][ml * 16 + ka + 1];
    agi0[ks].x = gst[wv][1][ml * 16 + ka];  agi0[ks].y = gst[wv][1][ml * 16 + ka + 1];
    agr1[ks].x = gst[wv][2][ml * 16 + ka];  agr1[ks].y = gst[wv][2][ml * 16 + ka + 1];
    agi1[ks].x = gst[wv][3][ml * 16 + ka];  agi1[ks].y = gst[wv][3][ml * 16 + ka + 1];
  }

  // ---- out = Gr*C2 + Gi*S2 over 16 N-tiles, B-frags shared by both tiles ----
  for (int j = 0; j < 16; ++j) {
    const int wc = j * 16 + ml;                // w column for B frag
    v8f o0 = {}, o1 = {};
    int iw = wc * 2 * half;                    // running (kw * w) mod 256
#pragma unroll
    for (int ks = 0; ks < 4; ++ks) {
      const int i0 = iw & 255;
      const int i1 = (iw + wc) & 255;
      v2f bc, bs;
      bc.x = c2[i0];  bc.y = c2[i1];
      bs.x = s2[i0];  bs.y = s2[i1];
      o0 = wmma4(agr0[ks], bc, o0);  o0 = wmma4(agi0[ks], bs, o0);
      o1 = wmma4(agr1[ks], bc, o1);  o1 = wmma4(agi1[ks], bs, o1);
      iw += 4 * wc;
    }
#pragma unroll
    for (int v = 0; v < 8; ++v) {
      const int r = v + 8 * half;
      outim[(size_t)(wv * 32 + r) * 256 + j * 16 + ml]      = o0[v];
      outim[(size_t)(wv * 32 + 16 + r) * 256 + j * 16 + ml] = o1[v];
    }
  }
}

// ---------------------------------------------------------------------------
extern "C" void kernel_launch(void* const* d_in, const int* in_sizes, int n_in,
                              void* d_out, int out_size, void* d_ws, size_t ws_size,
                              hipStream_t stream) {
  const float* x   = (const float*)d_in[0];   // [16,32,256,256]
  const float* wts = (const float*)d_in[1];   // [32,32,16,16,2]
  float* out = (float*)d_out;                 // [16,32,256,256]

  float* xft = (float*)d_ws;                  // 512 imgs * 512 floats = 1 MB
  float* low = xft + 512 * 512;               // 1 MB

  hipLaunchKernelGGL(fno_fwd_dft, dim3(512), dim3(256), 0, stream, x, xft);
  hipLaunchKernelGGL(fno_mix,     dim3(256), dim3(32),  0, stream, xft, wts, low);
  hipLaunchKernelGGL(fno_inv,     dim3(512), dim3(256), 0, stream, low, out);
}